// baseline_lstm_42064909697680
// MI455X (gfx1250) — compile-verified
//
#include <hip/hip_runtime.h>
#include <hip/hip_bf16.h>
#include <math.h>

// ---------------------------------------------------------------------------
// Shapes (from the reference)
// ---------------------------------------------------------------------------
#define EMBED   128
#define DEC_HID 66
#define HP      68          // hidden padded to multiple of 4 (K granularity)
#define GATES   264         // 4 * DEC_HID
#define GP      272         // gates padded to 17 tiles of 16
#define NWAVE   17          // one wave per 16-wide gate-column tile
#define NTHREADS (NWAVE * 32)

typedef __attribute__((ext_vector_type(2))) float v2f;
typedef __attribute__((ext_vector_type(8))) float v8f;

__device__ __forceinline__ float sigmoidf_(float x) {
    return 1.0f / (1.0f + expf(-x));
}

__device__ __forceinline__ v8f wmma4(v2f a, v2f b, v8f c) {
    return __builtin_amdgcn_wmma_f32_16x16x4_f32(
        false, a, false, b, (short)0, c, false, false);
}

// ---------------------------------------------------------------------------
// Persistent decoder recurrence. One workgroup, 17 waves.
// Weights live in VGPRs as WMMA B-fragments for the whole 512-step scan.
// All batch rows of the reference are identical (h0=c0=0, x0=emb[0] for every
// row), so we run one M=16 replicated tile and broadcast later.
// Independent WMMA accumulator chains (4 for gates, 2 for fc) to cut the per-step
// serial D->C dependency chain (the recurrence is pure latency).
// ---------------------------------------------------------------------------
__global__ __launch_bounds__(NTHREADS) void decoder_persistent_kernel(
    const float* __restrict__ emb,      // [VOCAB,128]  (row 0 used)
    const float* __restrict__ W_ih,     // [264,128]
    const float* __restrict__ W_hh,     // [264,66]
    const float* __restrict__ b_ih,     // [264]
    const float* __restrict__ b_hh,     // [264]
    const float* __restrict__ fc_W,     // [128,66]
    const float* __restrict__ fc_b,     // [128]
    float* __restrict__ hseq,           // [T,66] scratch out
    int T)
{
    __shared__ float Xs[16 * EMBED];    // decoder input x_in, 16 replicated rows
    __shared__ float Hs[16 * HP];       // hidden state (cols 66..67 stay 0)
    __shared__ float Cs[16 * DEC_HID];  // cell state
    __shared__ float Gs[16 * GP];       // pre-activation gates
    __shared__ float dbs[GATES];        // b_ih + b_hh
    __shared__ float fcbs[EMBED];       // fc bias

    const int tid  = threadIdx.x;
    const int w    = tid >> 5;          // wave id 0..16
    const int lane = tid & 31;
    const int col  = lane & 15;         // N index / A-row index
    const int hi   = lane >> 4;         // 0: K pair {0,1}; 1: K pair {2,3}
    const int kb   = hi * 2;
    const int g0   = w * 16;            // this wave's gate-column tile base
    const int gcol = g0 + col;

    // ---- Load per-wave weight B-fragments into registers (kept for all T) ----
    // B 4x16 fp32 layout: VGPR0 = B[k0+2*hi][col], VGPR1 = B[k0+2*hi+1][col]
    v2f fih[32];                                // W_ih^T slice, K = 0..127
    #pragma unroll
    for (int kc = 0; kc < 32; ++kc) {
        int k = kc * 4 + kb;
        float x = 0.0f, y = 0.0f;
        if (gcol < GATES) {
            x = W_ih[gcol * EMBED + k];
            y = W_ih[gcol * EMBED + k + 1];
        }
        v2f f; f.x = x; f.y = y; fih[kc] = f;
    }
    v2f fhh[17];                                // W_hh^T slice, K = 0..67 (padded)
    #pragma unroll
    for (int kc = 0; kc < 17; ++kc) {
        int k = kc * 4 + kb;
        float x = 0.0f, y = 0.0f;
        if (gcol < GATES && k     < DEC_HID) x = W_hh[gcol * DEC_HID + k];
        if (gcol < GATES && k + 1 < DEC_HID) y = W_hh[gcol * DEC_HID + k + 1];
        v2f f; f.x = x; f.y = y; fhh[kc] = f;
    }
    v2f ffc[17];                                // fc_W^T slice (waves 0..7 only)
    #pragma unroll
    for (int kc = 0; kc < 17; ++kc) {
        int k = kc * 4 + kb;
        float x = 0.0f, y = 0.0f;
        if (w < 8) {
            int n = g0 + col;                   // embed column 0..127
            if (k     < DEC_HID) x = fc_W[n * DEC_HID + k];
            if (k + 1 < DEC_HID) y = fc_W[n * DEC_HID + k + 1];
        }
        v2f f; f.x = x; f.y = y; ffc[kc] = f;
    }

    // ---- Initialize LDS state ----
    for (int i = tid; i < 16 * EMBED;   i += NTHREADS) Xs[i] = emb[i & (EMBED - 1)]; // emb[0] replicated
    for (int i = tid; i < 16 * HP;      i += NTHREADS) Hs[i] = 0.0f;
    for (int i = tid; i < 16 * DEC_HID; i += NTHREADS) Cs[i] = 0.0f;
    for (int i = tid; i < GATES;        i += NTHREADS) dbs[i] = b_ih[i] + b_hh[i];
    for (int i = tid; i < EMBED;        i += NTHREADS) fcbs[i] = fc_b[i];
    __syncthreads();

    const int row = col;   // A-matrix row held by this lane (M = lane & 15)

    for (int t = 0; t < T; ++t) {
        // -------- G = X @ W_ih^T + H @ W_hh^T  (this wave's 16 gate cols) ----
        // 4 independent accumulator chains (16/16/9/8 WMMAs) for ILP.
        v8f acc0 = {}, acc1 = {}, acc2 = {}, acc3 = {};
        #pragma unroll
        for (int kc = 0; kc < 32; kc += 2) {
            int k0 = kc * 4 + kb;
            int k1 = k0 + 4;
            v2f a0; a0.x = Xs[row * EMBED + k0]; a0.y = Xs[row * EMBED + k0 + 1];
            v2f a1; a1.x = Xs[row * EMBED + k1]; a1.y = Xs[row * EMBED + k1 + 1];
            acc0 = wmma4(a0, fih[kc],     acc0);
            acc1 = wmma4(a1, fih[kc + 1], acc1);
        }
        #pragma unroll
        for (int kc = 0; kc < 16; kc += 2) {
            int k0 = kc * 4 + kb;
            int k1 = k0 + 4;
            v2f a0; a0.x = Hs[row * HP + k0]; a0.y = Hs[row * HP + k0 + 1];
            v2f a1; a1.x = Hs[row * HP + k1]; a1.y = Hs[row * HP + k1 + 1];
            acc2 = wmma4(a0, fhh[kc],     acc2);
            acc3 = wmma4(a1, fhh[kc + 1], acc3);
        }
        {
            int k = 16 * 4 + kb;
            v2f a; a.x = Hs[row * HP + k]; a.y = Hs[row * HP + k + 1];
            acc2 = wmma4(a, fhh[16], acc2);
        }
        // C/D layout: element v -> M = v + 8*hi, N = col
        #pragma unroll
        for (int v = 0; v < 8; ++v)
            Gs[(v + 8 * hi) * GP + g0 + col] =
                (acc0[v] + acc1[v]) + (acc2[v] + acc3[v]);
        __syncthreads();

        // -------- elementwise LSTM cell (torch gate order i,f,g,o) ----------
        for (int idx = tid; idx < 16 * DEC_HID; idx += NTHREADS) {
            int r = idx / DEC_HID;
            int j = idx - r * DEC_HID;
            const float* gr = &Gs[r * GP];
            float ig = sigmoidf_(gr[j]               + dbs[j]);
            float fg = sigmoidf_(gr[DEC_HID + j]     + dbs[DEC_HID + j]);
            float gg = tanhf    (gr[2 * DEC_HID + j] + dbs[2 * DEC_HID + j]);
            float og = sigmoidf_(gr[3 * DEC_HID + j] + dbs[3 * DEC_HID + j]);
            float cv = fg * Cs[idx] + ig * gg;
            Cs[idx] = cv;
            float hv = og * tanhf(cv);
            Hs[r * HP + j] = hv;
            if (r == 0) hseq[t * DEC_HID + j] = hv;   // one canonical row
        }
        __syncthreads();

        // -------- X_next = H @ fc_W^T + fc_b  (waves 0..7, 8 tiles of 16) ---
        if (w < 8) {
            v8f fa0 = {}, fa1 = {};
            #pragma unroll
            for (int kc = 0; kc < 16; kc += 2) {
                int k0 = kc * 4 + kb;
                int k1 = k0 + 4;
                v2f a0; a0.x = Hs[row * HP + k0]; a0.y = Hs[row * HP + k0 + 1];
                v2f a1; a1.x = Hs[row * HP + k1]; a1.y = Hs[row * HP + k1 + 1];
                fa0 = wmma4(a0, ffc[kc],     fa0);
                fa1 = wmma4(a1, ffc[kc + 1], fa1);
            }
            {
                int k = 16 * 4 + kb;
                v2f a; a.x = Hs[row * HP + k]; a.y = Hs[row * HP + k + 1];
                fa0 = wmma4(a, ffc[16], fa0);
            }
            #pragma unroll
            for (int v = 0; v < 8; ++v)
                Xs[(v + 8 * hi) * EMBED + g0 + col] =
                    fa0[v] + fa1[v] + fcbs[g0 + col];
        }
        __syncthreads();
    }
}

// ---------------------------------------------------------------------------
// Broadcast the single trajectory to all 128 identical batch rows.
// Pure bandwidth: 17.3 MB of stores (~0.74 us at 23.3 TB/s); hseq hits L2.
// ---------------------------------------------------------------------------
__global__ void broadcast_kernel(const float* __restrict__ hseq,
                                 float* __restrict__ out,
                                 int total, int TH)
{
    int i = blockIdx.x * blockDim.x + threadIdx.x;
    if (i < total) {
        int th = i % TH;          // i = b*T*H + (t*H + j)
        out[i] = hseq[th];
    }
}

extern "C" void kernel_launch(void* const* d_in, const int* in_sizes, int n_in,
                              void* d_out, int out_size, void* d_ws, size_t ws_size,
                              hipStream_t stream) {
    (void)n_in; (void)out_size; (void)ws_size;
    // setup_inputs order:
    // 0:x 1:c 2:emb 3:enc_W_ih 4:enc_W_hh 5:enc_b_ih 6:enc_b_hh
    // 7:dec_W_ih 8:dec_W_hh 9:dec_b_ih 10:dec_b_hh 11:fc_W 12:fc_b
    const float* emb    = (const float*)d_in[2];
    const float* dW_ih  = (const float*)d_in[7];
    const float* dW_hh  = (const float*)d_in[8];
    const float* db_ih  = (const float*)d_in[9];
    const float* db_hh  = (const float*)d_in[10];
    const float* fc_W   = (const float*)d_in[11];
    const float* fc_b   = (const float*)d_in[12];

    const int B = 128;
    const int T = in_sizes[1] / (B * DEC_HID);   // c is [B,T,66] -> T = 512

    float* hseq = (float*)d_ws;                  // [T,66] scratch

    decoder_persistent_kernel<<<1, NTHREADS, 0, stream>>>(
        emb, dW_ih, dW_hh, db_ih, db_hh, fc_W, fc_b, hseq, T);

    const int total = B * T * DEC_HID;
    broadcast_kernel<<<(total + 255) / 256, 256, 0, stream>>>(
        hseq, (float*)d_out, total, T * DEC_HID);
}